// SupConLoss_49503793054189
// MI455X (gfx1250) — compile-verified
//
#include <hip/hip_runtime.h>
#include <hip/hip_bf16.h>
#include <math.h>

// SupCon loss on gfx1250 using V_WMMA_F32_16X16X4_F32 for the Gram matrix.
//
// ws layout (floats): anchor[M*32] | rowmax[M] | negsum[M] | possum[M]

typedef __attribute__((ext_vector_type(2))) float v2f;
typedef __attribute__((ext_vector_type(8))) float v8f;

#define SC_TEMP      0.07f
#define SC_BASETEMP  0.07f
#define SC_INVTEMP   (1.0f / SC_TEMP)
#define SC_D         32          // feature dim (reference: D=32)
#define SC_B         2           // batch (reference: B=2)

// ---------------------------------------------------------------------------
// Kernel 0: gather anchor[M][32] from features (B,D,H,W,Z) via gather_idx
// anchor row m = v*C + c ; source row n = gather_idx[c, v]
// feats[n][d] = features[(b*D + d)*HWZ + (n % HWZ)] with b = n / HWZ
// ---------------------------------------------------------------------------
__global__ void sc_gather(const float* __restrict__ feat,
                          const int* __restrict__ gidx,
                          float* __restrict__ anchor,
                          int C, int n_view, int HWZ, int total) {
  int t = blockIdx.x * blockDim.x + threadIdx.x;
  if (t >= total) return;
  int m = t >> 5;          // / 32
  int d = t & 31;
  int v = m / C;
  int c = m - v * C;
  int n = gidx[c * n_view + v];
  int b = n / HWZ;
  int rem = n - b * HWZ;
  anchor[t] = feat[(size_t)(b * SC_D + d) * (size_t)HWZ + (size_t)rem];
}

// 16x16 logits tile via 8x V_WMMA_F32_16X16X4_F32 (K = 32)
__device__ __forceinline__ v8f sc_tile(const v2f a[8],
                                       const float* __restrict__ bRow,
                                       int koff) {
  v8f c = {};
#pragma unroll
  for (int q = 0; q < 8; ++q) {
    v2f b = *(const v2f*)(bRow + q * 4 + koff);
    c = __builtin_amdgcn_wmma_f32_16x16x4_f32(
        /*neg_a=*/false, a[q], /*neg_b=*/false, b,
        /*c_mod=*/(short)0, c, /*reuse_a=*/false, /*reuse_b=*/false);
  }
  return c;
}

// ---------------------------------------------------------------------------
// Pass 1: rowmax[i] = max_j (a_i . a_j) / TEMP          (one wave per 16 rows)
// ---------------------------------------------------------------------------
__global__ void sc_rowmax(const float* __restrict__ anchor,
                          float* __restrict__ rowmax, int M) {
  const int lane = threadIdx.x;
  const int l15 = lane & 15;
  const int h = lane >> 4;
  const int koff = 2 * h;
  const int blk = blockIdx.x;
  const int rowA = blk * 16 + l15;

  v2f a[8];
#pragma unroll
  for (int q = 0; q < 8; ++q)
    a[q] = *(const v2f*)(anchor + rowA * 32 + q * 4 + koff);

  float mx[8];
#pragma unroll
  for (int r = 0; r < 8; ++r) mx[r] = -3.402823466e+38f;

  const int ntiles = M >> 4;
  for (int j = 0; j < ntiles; ++j) {
    const float* bRow = anchor + (j * 16 + l15) * 32;
    if (j + 1 < ntiles) __builtin_prefetch(bRow + 16 * 32, 0, 0);
    v8f c = sc_tile(a, bRow, koff);
#pragma unroll
    for (int r = 0; r < 8; ++r) mx[r] = fmaxf(mx[r], c[r] * SC_INVTEMP);
  }
#pragma unroll
  for (int r = 0; r < 8; ++r) {
    float v = mx[r];
    for (int s = 1; s < 16; s <<= 1) v = fmaxf(v, __shfl_xor(v, s, 32));
    mx[r] = v;
  }
  if (l15 == 0) {
#pragma unroll
    for (int r = 0; r < 8; ++r) rowmax[blk * 16 + 8 * h + r] = mx[r];
  }
}

// ---------------------------------------------------------------------------
// Pass 2: negsum[i] = sum_{j: cls(j)!=cls(i)} exp(logit_ij - rowmax[i])
// ---------------------------------------------------------------------------
__global__ void sc_negsum(const float* __restrict__ anchor,
                          const float* __restrict__ rowmax,
                          float* __restrict__ negsum, int M, int C) {
  const int lane = threadIdx.x;
  const int l15 = lane & 15;
  const int h = lane >> 4;
  const int koff = 2 * h;
  const int blk = blockIdx.x;
  const int rowA = blk * 16 + l15;

  v2f a[8];
#pragma unroll
  for (int q = 0; q < 8; ++q)
    a[q] = *(const v2f*)(anchor + rowA * 32 + q * 4 + koff);

  float rmax[8], acc[8];
  int mycls[8];
#pragma unroll
  for (int r = 0; r < 8; ++r) {
    int i = blk * 16 + 8 * h + r;
    rmax[r] = rowmax[i];
    mycls[r] = i % C;
    acc[r] = 0.0f;
  }

  const int ntiles = M >> 4;
  for (int j = 0; j < ntiles; ++j) {
    const int jcol = j * 16 + l15;
    const int jc = jcol % C;
    const float* bRow = anchor + jcol * 32;
    if (j + 1 < ntiles) __builtin_prefetch(bRow + 16 * 32, 0, 0);
    v8f c = sc_tile(a, bRow, koff);
#pragma unroll
    for (int r = 0; r < 8; ++r) {
      float e = __expf(c[r] * SC_INVTEMP - rmax[r]);
      acc[r] += (jc != mycls[r]) ? e : 0.0f;
    }
  }
#pragma unroll
  for (int r = 0; r < 8; ++r) {
    float v = acc[r];
    for (int s = 1; s < 16; s <<= 1) v += __shfl_xor(v, s, 32);
    acc[r] = v;
  }
  if (l15 == 0) {
#pragma unroll
    for (int r = 0; r < 8; ++r) negsum[blk * 16 + 8 * h + r] = acc[r];
  }
}

// ---------------------------------------------------------------------------
// Pass 3: possum[i] = sum_{j: cls(j)==cls(i), j!=i} z - log(exp(z) + negsum[i])
//         with z = logit_ij - rowmax[i]
// ---------------------------------------------------------------------------
__global__ void sc_possum(const float* __restrict__ anchor,
                          const float* __restrict__ rowmax,
                          const float* __restrict__ negsum,
                          float* __restrict__ possum, int M, int C) {
  const int lane = threadIdx.x;
  const int l15 = lane & 15;
  const int h = lane >> 4;
  const int koff = 2 * h;
  const int blk = blockIdx.x;
  const int rowA = blk * 16 + l15;

  v2f a[8];
#pragma unroll
  for (int q = 0; q < 8; ++q)
    a[q] = *(const v2f*)(anchor + rowA * 32 + q * 4 + koff);

  float rmax[8], nsum[8], acc[8];
  int irow[8], mycls[8];
#pragma unroll
  for (int r = 0; r < 8; ++r) {
    int i = blk * 16 + 8 * h + r;
    irow[r] = i;
    rmax[r] = rowmax[i];
    nsum[r] = negsum[i];
    mycls[r] = i % C;
    acc[r] = 0.0f;
  }

  const int ntiles = M >> 4;
  for (int j = 0; j < ntiles; ++j) {
    const int jcol = j * 16 + l15;
    const int jc = jcol % C;
    const float* bRow = anchor + jcol * 32;
    if (j + 1 < ntiles) __builtin_prefetch(bRow + 16 * 32, 0, 0);
    v8f c = sc_tile(a, bRow, koff);
#pragma unroll
    for (int r = 0; r < 8; ++r) {
      float z = c[r] * SC_INVTEMP - rmax[r];
      float lp = z - __logf(__expf(z) + nsum[r]);
      bool pos = (jc == mycls[r]) && (jcol != irow[r]);
      acc[r] += pos ? lp : 0.0f;
    }
  }
#pragma unroll
  for (int r = 0; r < 8; ++r) {
    float v = acc[r];
    for (int s = 1; s < 16; s <<= 1) v += __shfl_xor(v, s, 32);
    acc[r] = v;
  }
  if (l15 == 0) {
#pragma unroll
    for (int r = 0; r < 8; ++r) possum[blk * 16 + 8 * h + r] = acc[r];
  }
}

// ---------------------------------------------------------------------------
// Final: loss = (TEMP/BASE_TEMP) * mean_i( -possum[i] / cnt_i )
// cnt_i = n_view * #{c': y[c'] == y[i%C]} - 1
// ---------------------------------------------------------------------------
__global__ void sc_finalize(const float* __restrict__ possum,
                            const float* __restrict__ y,
                            float* __restrict__ out,
                            int M, int C, int n_view) {
  __shared__ float sdata[256];
  const int tid = threadIdx.x;
  float local = 0.0f;
  for (int i = tid; i < M; i += blockDim.x) {
    int c = i % C;
    int same = 0;
    for (int cc = 0; cc < C; ++cc) same += (y[cc] == y[c]) ? 1 : 0;
    float cnt = (float)(n_view * same - 1);
    local += -possum[i] / cnt;
  }
  sdata[tid] = local;
  __syncthreads();
  for (int s = 128; s > 0; s >>= 1) {
    if (tid < s) sdata[tid] += sdata[tid + s];
    __syncthreads();
  }
  if (tid == 0) out[0] = (SC_TEMP / SC_BASETEMP) * sdata[0] / (float)M;
}

extern "C" void kernel_launch(void* const* d_in, const int* in_sizes, int n_in,
                              void* d_out, int out_size, void* d_ws, size_t ws_size,
                              hipStream_t stream) {
  const float* feat = (const float*)d_in[0];   // [B, D, H, W, Z] f32
  const int* gidx = (const int*)d_in[1];       // [C, n_view] i32
  const float* yanc = (const float*)d_in[2];   // [C] f32

  const int C = in_sizes[2];
  const int n_view = in_sizes[1] / C;
  const int M = C * n_view;                    // 4096
  const int N = in_sizes[0] / SC_D;            // B*H*W*Z
  const int HWZ = N / SC_B;

  float* ws = (float*)d_ws;
  float* anchor = ws;                          // M*32
  float* rowmax = anchor + (size_t)M * 32;     // M
  float* negsum = rowmax + M;                  // M
  float* possum = negsum + M;                  // M

  const int total = M * SC_D;
  sc_gather<<<(total + 255) / 256, 256, 0, stream>>>(feat, gidx, anchor,
                                                     C, n_view, HWZ, total);

  const int nblk = M / 16;                     // one wave per 16-row block
  sc_rowmax<<<nblk, 32, 0, stream>>>(anchor, rowmax, M);
  sc_negsum<<<nblk, 32, 0, stream>>>(anchor, rowmax, negsum, M, C);
  sc_possum<<<nblk, 32, 0, stream>>>(anchor, rowmax, negsum, possum, M, C);
  sc_finalize<<<1, 256, 0, stream>>>(possum, yanc, (float*)d_out, M, C, n_view);
}